// beam_search_decoder_45810121179741
// MI455X (gfx1250) — compile-verified
//
#include <hip/hip_runtime.h>
#include <hip/hip_bf16.h>
#include <stdint.h>
#include <math.h>

// ---------------------------------------------------------------------------
// Beam-search decoder (K=3, BLANK=0) for emission [T=2048, V=32000] f32.
// Kernel 1: per-row top-3 via async global->LDS double-buffered streaming
//           (gfx1250 global_load_async_to_lds_b128 + s_wait_asynccnt /
//           s_wait_dscnt). Branchless top-3 using a monotone 64-bit key:
//           key = (float_bits << 32) | ~idx   (values in (0,1] => positive
//           bit patterns, order-isomorphic), exactly lax.top_k order
//           (value desc, index asc). LDS tree merge, store log(topv)+topi.
// Kernel 2: serial 9->top3 scan (pruned to 6 candidates), backtrack beam 1,
//           emit seq (as f32), keep mask (0/1 f32), final 3 scores.
// Output layout: d_out[0..T) = seq, [T..2T) = keep, [2T..2T+3) = scores.
// ---------------------------------------------------------------------------

#define VOCAB 32000
#define TMAX  2048

// branchless insert of 64-bit key into sorted-desc 3-slot list (s0k>=s1k>=s2k)
// monotone: g0 => g1 => g2, so the select chain is exact.
#define INSK(k_)                                                      \
  do {                                                                \
    unsigned long long _k = (k_);                                     \
    const bool g0 = _k > s0k;                                         \
    const bool g1 = _k > s1k;                                         \
    const bool g2 = _k > s2k;                                         \
    s2k = g1 ? s1k : (g2 ? _k : s2k);                                 \
    s1k = g0 ? s0k : (g1 ? _k : s1k);                                 \
    s0k = g0 ? _k : s0k;                                              \
  } while (0)

__device__ __forceinline__ unsigned long long mk_key(float v, int idx) {
  return ((unsigned long long)__float_as_uint(v) << 32) |
         (unsigned int)(~idx);
}

__device__ __forceinline__ void async_to_lds_b128(uint32_t lds_off,
                                                  const void* gptr) {
  unsigned long long ga = (unsigned long long)(uintptr_t)gptr;
  asm volatile("global_load_async_to_lds_b128 %0, %1, off"
               :: "v"(lds_off), "v"(ga) : "memory");
}

__global__ __launch_bounds__(256) void topk_rows_kernel(
    const float* __restrict__ em, float* __restrict__ lv,
    int* __restrict__ ti) {
  // staging: 2 buffers x 1024 floats (256 threads x float4)
  __shared__ __align__(16) float stage[2][1024];
  __shared__ unsigned long long rk[3 * 256];

  const int tid = threadIdx.x;
  const int row_id = blockIdx.x;
  const float* row = em + (size_t)row_id * VOCAB;

  // sentinel key 0 < any real key (values > 0 => fbits > 0)
  unsigned long long s0k = 0ull, s1k = 0ull, s2k = 0ull;

  // LDS byte offset of this thread's 16B slot in buffer 0
  const uint32_t lds0 =
      (uint32_t)(uintptr_t)(&stage[0][0]) + (uint32_t)(tid * 16);

  // 31 full chunks of 1024 floats (elements 0..31743), async double-buffered.
  // Each wave only reads back the LDS region its own lanes copied, so
  // per-wave ASYNCcnt waits suffice (no workgroup barrier in the loop).
  async_to_lds_b128(lds0, row + tid * 4);  // chunk 0 -> buf 0
#pragma unroll 1
  for (int c = 0; c < 31; ++c) {
    const int buf = c & 1;
    if (c + 1 < 31) {
      async_to_lds_b128(lds0 + (uint32_t)((buf ^ 1) * 4096),
                        row + (c + 1) * 1024 + tid * 4);
      asm volatile("s_wait_asynccnt 1" ::: "memory");  // chunk c landed
    } else {
      asm volatile("s_wait_asynccnt 0" ::: "memory");
    }
    float4 d = *(const float4*)(&stage[buf][tid * 4]);
    // fence the LDS read before next iteration's async overwrites this buffer
    asm volatile("s_wait_dscnt 0" ::: "memory");
    const int e = c * 1024 + tid * 4;
    INSK(mk_key(d.x, e + 0));
    INSK(mk_key(d.y, e + 1));
    INSK(mk_key(d.z, e + 2));
    INSK(mk_key(d.w, e + 3));
  }

  // tail: elements 31744..31999 (256 floats), direct coalesced loads
  if (tid < 64) {
    const int e = 31744 + tid * 4;
    float4 d = *(const float4*)(row + e);
    INSK(mk_key(d.x, e + 0));
    INSK(mk_key(d.y, e + 1));
    INSK(mk_key(d.z, e + 2));
    INSK(mk_key(d.w, e + 3));
  }

  // LDS tree merge of 256 per-thread top-3 key lists
  rk[tid] = s0k; rk[256 + tid] = s1k; rk[512 + tid] = s2k;
  __syncthreads();
  for (int s = 128; s > 0; s >>= 1) {
    if (tid < s) {
      const int o = tid + s;
      const unsigned long long ok0 = rk[o];
      const unsigned long long ok1 = rk[256 + o];
      const unsigned long long ok2 = rk[512 + o];
      INSK(ok0);
      INSK(ok1);
      INSK(ok2);
      rk[tid] = s0k; rk[256 + tid] = s1k; rk[512 + tid] = s2k;
    }
    __syncthreads();
  }

  if (tid == 0) {
    lv[3 * row_id + 0] = logf(__uint_as_float((unsigned int)(s0k >> 32)));
    lv[3 * row_id + 1] = logf(__uint_as_float((unsigned int)(s1k >> 32)));
    lv[3 * row_id + 2] = logf(__uint_as_float((unsigned int)(s2k >> 32)));
    ti[3 * row_id + 0] = (int)~(unsigned int)s0k;
    ti[3 * row_id + 1] = (int)~(unsigned int)s1k;
    ti[3 * row_id + 2] = (int)~(unsigned int)s2k;
  }
}

// branchless candidate insert for the scan: comparator (value desc, flat-
// index asc); slots totally ordered => g0 => g1 => g2 monotone.
#define INSF(val_, ff_)                                                    \
  do {                                                                     \
    float _v = (val_); int _f = (ff_);                                     \
    const bool g0 = (_v > w0) || ((_v == w0) && (_f < f0));                \
    const bool g1 = (_v > w1) || ((_v == w1) && (_f < f1));                \
    const bool g2 = (_v > w2) || ((_v == w2) && (_f < f2));                \
    w2 = g1 ? w1 : (g2 ? _v : w2); f2 = g1 ? f1 : (g2 ? _f : f2);          \
    w1 = g0 ? w0 : (g1 ? _v : w1); f1 = g0 ? f0 : (g1 ? _f : f1);          \
    w0 = g0 ? _v : w0;             f0 = g0 ? _f : f0;                      \
  } while (0)

__global__ __launch_bounds__(256) void beam_scan_kernel(
    const float* __restrict__ lv, const int* __restrict__ ti,
    float* __restrict__ out, int T) {
  __shared__ float Llds[3 * TMAX];
  __shared__ unsigned char pb[TMAX];    // packed parents (2 bits x 3 beams)
  __shared__ unsigned char qb[TMAX];    // packed token slot (best_f % 3)
  __shared__ unsigned char bsel[TMAX];  // beam lineage of final beam 1
  __shared__ int seq[TMAX];

  const int tid = threadIdx.x;

  // stage log-top3 triples into LDS (24 KB)
  for (int k = tid; k < 3 * T; k += blockDim.x) Llds[k] = lv[k];
  __syncthreads();

  if (tid == 0) {
    // scores init: [0, -inf, -inf]
    float s0 = 0.0f, s1 = -INFINITY, s2 = -INFINITY;
    for (int t = 0; t < T; ++t) {
      const float L0 = Llds[3 * t + 0];
      const float L1 = Llds[3 * t + 1];
      const float L2 = Llds[3 * t + 2];
      // 3x3 candidate matrix has sorted rows/cols; only flat indices
      // {0,1,2,3,4,6} can reach the top-3 under (value desc, index asc).
      float w0 = -INFINITY, w1 = -INFINITY, w2 = -INFINITY;
      int   f0 = 9, f1 = 9, f2 = 9;
      INSF(s0 + L0, 0);
      INSF(s0 + L1, 1);
      INSF(s0 + L2, 2);
      INSF(s1 + L0, 3);
      INSF(s1 + L1, 4);
      INSF(s2 + L0, 6);
      const int p0 = f0 / 3, p1 = f1 / 3, p2 = f2 / 3;
      const int q0 = f0 % 3, q1 = f1 % 3, q2 = f2 % 3;
      pb[t] = (unsigned char)(p0 | (p1 << 2) | (p2 << 4));
      qb[t] = (unsigned char)(q0 | (q1 << 2) | (q2 << 4));
      s0 = w0; s1 = w1; s2 = w2;
    }
    out[2 * T + 0] = s0;
    out[2 * T + 1] = s1;
    out[2 * T + 2] = s2;
    // backtrack beam 1 (the SECOND-best beam): ~3 dependent VALU ops/step
    int b = 1;
    for (int t = T - 1; t >= 0; --t) {
      bsel[t] = (unsigned char)b;
      b = (pb[t] >> (2 * b)) & 3;
    }
  }
  __syncthreads();

  // gather tokens of the selected lineage (parallel)
  for (int t = tid; t < T; t += blockDim.x) {
    const int b = bsel[t];
    const int q = (qb[t] >> (2 * b)) & 3;
    seq[t] = ti[3 * t + q];
  }
  __syncthreads();

  // unique_consecutive "change" mask & blank drop (parallel)
  for (int t = tid; t < T; t += blockDim.x) {
    const int s = seq[t];
    const bool change = (t == 0) || (s != seq[t - 1]);
    out[t]     = (float)s;
    out[T + t] = (change && (s != 0)) ? 1.0f : 0.0f;
  }
}

extern "C" void kernel_launch(void* const* d_in, const int* in_sizes, int n_in,
                              void* d_out, int out_size, void* d_ws,
                              size_t ws_size, hipStream_t stream) {
  const float* em = (const float*)d_in[0];
  const int T = in_sizes[0] / VOCAB;  // 2048

  float* lv = (float*)d_ws;                                         // T*3 f32
  int*   ti = (int*)((char*)d_ws + (size_t)T * 3 * sizeof(float));  // T*3 i32

  topk_rows_kernel<<<T, 256, 0, stream>>>(em, lv, ti);
  beam_scan_kernel<<<1, 256, 0, stream>>>(lv, ti, (float*)d_out, T);
}